// JensenShannonDivergenceWeightedScaled_21517786153521
// MI455X (gfx1250) — compile-verified
//
#include <hip/hip_runtime.h>
#include <math.h>

// ---------------------------------------------------------------------------
// JS divergence (weighted, scaled), B=1024 rows, T=8192 positions, 3 classes.
// HBM-bound: ~133 MB traffic -> ~5.7 us at 23.3 TB/s. One WG (8 wave32) per
// row; b128-vectorized streaming; WMMA f32 pipe used for the final reduce.
// ---------------------------------------------------------------------------

typedef float v2f __attribute__((ext_vector_type(2)));
typedef float v8f __attribute__((ext_vector_type(8)));

#define TDIM 8192
#define BROWS 1024
#define NTHREADS 256
#define VEC 4
#define LOG_EPS  (-16.118095651f)        // logf(1e-7f)
#define LN2F     (0.69314718056f)
#define JS_SCALE (2.885390081777927f)    // -1/((1-0.5)*ln(0.5)) = 2/ln2

// Per-position weighted JS divergence term (weights 0.5/0.5), y in {0,1,2}.
__device__ __forceinline__ float js_pos(float x0, float x1, float x2, int y)
{
    const float m  = fmaxf(x0, fmaxf(x1, x2));
    const float t0 = x0 - m, t1 = x1 - m, t2 = x2 - m;
    const float e0 = expf(t0), e1 = expf(t1), e2 = expf(t2);
    const float s  = e0 + e1 + e2;
    const float inv = 1.0f / s;
    const float ls  = logf(s);

    const float pr0 = e0 * inv, pr1 = e1 * inv, pr2 = e2 * inv;
    const float lp0 = t0 - ls,  lp1 = t1 - ls,  lp2 = t2 - ls;   // log p_c

    const float py  = (y == 0) ? pr0 : ((y == 1) ? pr1 : pr2);
    const float lmy = log1pf(py) - LN2F;          // log(0.5*(1+p_y))

    // log(clip(p_c, eps)) and log(clip(0.5*p_c, eps)); upper clip never hit
    const float la0 = fmaxf(lp0, LOG_EPS);
    const float la1 = fmaxf(lp1, LOG_EPS);
    const float la2 = fmaxf(lp2, LOG_EPS);
    const float lm0 = (y == 0) ? lmy : fmaxf(lp0 - LN2F, LOG_EPS);
    const float lm1 = (y == 1) ? lmy : fmaxf(lp1 - LN2F, LOG_EPS);
    const float lm2 = (y == 2) ? lmy : fmaxf(lp2 - LN2F, LOG_EPS);

    const float klp = pr0 * (la0 - lm0) + pr1 * (la1 - lm1) + pr2 * (la2 - lm2);
    // 0.5*KL(onehot||mean) + 0.5*KL(p||mean) = 0.5*(klp - lmy)
    return 0.5f * (klp - lmy);
}

__global__ __launch_bounds__(NTHREADS)
void jsd_row_kernel(const float* __restrict__ pred,
                    const int*   __restrict__ labels,
                    float*       __restrict__ row_out)
{
    __shared__ int   s_j;
    __shared__ float s_red[NTHREADS];

    const int row = blockIdx.x;
    const int tid = threadIdx.x;

    if (tid == 0) s_j = TDIM - 1;          // default: no sentinel -> j = T-1
    __syncthreads();

    const int* __restrict__ lab = labels + (size_t)row * TDIM;

    // ---- Phase A: first index with label==3 (b128 loads, min-reduction) ----
    int localMin = 0x7FFFFFFF;
    for (int base = tid * VEC; base < TDIM; base += NTHREADS * VEC) {
        const int4 l = *(const int4*)(lab + base);
        if (l.x == 3) localMin = min(localMin, base + 0);
        if (l.y == 3) localMin = min(localMin, base + 1);
        if (l.z == 3) localMin = min(localMin, base + 2);
        if (l.w == 3) localMin = min(localMin, base + 3);
    }
    if (localMin != 0x7FFFFFFF) atomicMin(&s_j, localMin);
    __syncthreads();
    const int j = s_j;                     // valid positions are idx < j

    // ---- Phase B: stream logits with b128 loads, accumulate JS -------------
    const float* __restrict__ p0 = pred + (size_t)row * 3 * TDIM;
    const float* __restrict__ p1 = p0 + TDIM;
    const float* __restrict__ p2 = p1 + TDIM;

    float acc = 0.0f;
    for (int base = tid * VEC; base < j; base += NTHREADS * VEC) {
        // prefetch two iterations ahead on the three class streams
        const int pf = base + 2 * NTHREADS * VEC;
        if (pf < TDIM) {
            __builtin_prefetch(&p0[pf], 0, 0);
            __builtin_prefetch(&p1[pf], 0, 0);
            __builtin_prefetch(&p2[pf], 0, 0);
        }

        // base is a multiple of 4 and base < j <= T-1  =>  base+3 <= T-1:
        // the full 16B vectors are in-row; only contributions are masked.
        const float4 a0 = *(const float4*)(p0 + base);
        const float4 a1 = *(const float4*)(p1 + base);
        const float4 a2 = *(const float4*)(p2 + base);
        const int4   lv = *(const int4*)(lab + base);

        acc += js_pos(a0.x, a1.x, a2.x, lv.x);            // base+0 < j by loop cond
        if (base + 1 < j) acc += js_pos(a0.y, a1.y, a2.y, lv.y);
        if (base + 2 < j) acc += js_pos(a0.z, a1.z, a2.z, lv.z);
        if (base + 3 < j) acc += js_pos(a0.w, a1.w, a2.w, lv.w);
    }

    // ---- Block tree reduction (fixed order -> deterministic) ---------------
    s_red[tid] = acc;
    __syncthreads();
    #pragma unroll
    for (int off = NTHREADS / 2; off > 0; off >>= 1) {
        if (tid < off) s_red[tid] += s_red[tid + off];
        __syncthreads();
    }
    if (tid == 0) row_out[row] = s_red[0] / fmaxf((float)j, 1.0f);
}

// ---------------------------------------------------------------------------
// Final reduction of 1024 row values with one wave32, using the f32 WMMA pipe:
// A (16x4) = 32 lane partials (+zeros), B (4x16) = ones  =>  D rows = row-sums.
// ---------------------------------------------------------------------------
__global__ __launch_bounds__(32)
void jsd_final_kernel(const float* __restrict__ row_out,
                      float*       __restrict__ out)
{
    const int lane = threadIdx.x;

    float partial = 0.0f;
    for (int i = lane; i < BROWS; i += 32) partial += row_out[i];

    // A layout (16x4 f32): lanes 0-15 -> (M=lane, K=0|1); lanes 16-31 -> (M=lane-16, K=2|3)
    v2f a; a.x = partial; a.y = 0.0f;       // A[m,0]=partial_m, A[m,2]=partial_{m+16}
    v2f b; b.x = 1.0f;    b.y = 1.0f;       // B = all ones
    v8f c = {};
    // D[m,n] = sum_k A[m,k]*B[k,n] = partial_m + partial_{m+16}
    v8f d = __builtin_amdgcn_wmma_f32_16x16x4_f32(
        /*neg_a=*/false, a, /*neg_b=*/false, b,
        /*c_mod=*/(short)0, c, /*reuse_a=*/false, /*reuse_b=*/false);

    // lane (column n): VGPR r holds D[M = r + (lane>=16 ? 8:0), n]
    float t = d[0] + d[1] + d[2] + d[3] + d[4] + d[5] + d[6] + d[7];
    // column 0: lane 0 has sum of M=0..7, lane 16 has sum of M=8..15
    float total = __shfl(t, 0, 32) + __shfl(t, 16, 32);

    if (lane == 0) out[0] = total * (JS_SCALE / (float)BROWS);
}

extern "C" void kernel_launch(void* const* d_in, const int* in_sizes, int n_in,
                              void* d_out, int out_size, void* d_ws, size_t ws_size,
                              hipStream_t stream)
{
    const float* pred   = (const float*)d_in[0];   // (1024, 3, 8192) f32
    const int*   labels = (const int*)d_in[1];     // (1024, 8192) int
    float*       out    = (float*)d_out;           // scalar f32
    float*       rows   = (float*)d_ws;            // 1024 f32 scratch

    jsd_row_kernel<<<BROWS, NTHREADS, 0, stream>>>(pred, labels, rows);
    jsd_final_kernel<<<1, 32, 0, stream>>>(rows, out);
}